// QuantizedEmbedding_6743098655154
// MI455X (gfx1250) — compile-verified
//
#include <hip/hip_runtime.h>
#include <stdint.h>

// QuantizedEmbedding fused gather+dequant for MI455X (gfx1250).
// out[row, :] = clamp(rint(weights[tok, :]), -128, 127) * scales[tok]
// Pure bandwidth problem: ~128 MB of traffic -> ~5.5 us at 23.3 TB/s.

#define MODEL_DIM 2048
#define TPB       256                    // 8 wave32s per block
#define F_PER_T   (MODEL_DIM / TPB)      // 8 floats (= 32 bytes = 2x b128) per thread

typedef float v4f __attribute__((ext_vector_type(4)));

__device__ __forceinline__ float dq1(float w, float s) {
    float r = __builtin_rintf(w);                    // v_rndne_f32 (round half to even, matches jnp.round)
    r = fminf(fmaxf(r, -128.0f), 127.0f);            // clip to int8 grid
    return r * s;
}

__device__ __forceinline__ v4f dq4(v4f w, float s) {
    v4f r;
    r.x = dq1(w.x, s);
    r.y = dq1(w.y, s);
    r.z = dq1(w.z, s);
    r.w = dq1(w.w, s);
    return r;
}

// ---------------------------------------------------------------------------
// Path selection: async global->LDS DMA (gfx1250 ASYNCcnt path) if the
// toolchain declares the builtins, otherwise plain NT vector loads.
// ---------------------------------------------------------------------------
#if __has_builtin(__builtin_amdgcn_global_load_async_to_lds_b128)
#define USE_ASYNC_PATH 1
#endif

__device__ __forceinline__ void wait_async_0() {
#if __has_builtin(__builtin_amdgcn_s_wait_asynccnt)
    __builtin_amdgcn_s_wait_asynccnt(0);
#else
    asm volatile("s_wait_asynccnt 0x0" ::: "memory");
#endif
}
__device__ __forceinline__ void wait_async_2() {
#if __has_builtin(__builtin_amdgcn_s_wait_asynccnt)
    __builtin_amdgcn_s_wait_asynccnt(2);
#else
    asm volatile("s_wait_asynccnt 0x2" ::: "memory");
#endif
}

#ifdef USE_ASYNC_PATH
// Builtin signature (from the compiler diagnostic): param0 is
// 'int __attribute__((vector_size(16))) __device__ *' i.e. int4* in AS1
// (global); param1 is the LDS-side int4* (AS3). Args: (gsrc, ldst, imm
// offset applied to BOTH addresses, imm cpol).
typedef int v4i __attribute__((vector_size(16)));
typedef __attribute__((address_space(1))) v4i gv4i;   // global int4
typedef __attribute__((address_space(3))) v4i lv4i;   // LDS int4

// Per-thread: DMA 32 contiguous bytes of a weight row into this thread's LDS slice.
// Two async b128 issues per wave-instruction -> ASYNCcnt += 2 per wave per row.
__device__ __forceinline__ void async_row_slice_to_lds(const float* gsrc, float* ldst) {
    gv4i* g = (gv4i*)(uintptr_t)gsrc;
    lv4i* l = (lv4i*)(uintptr_t)ldst;   // low 32 bits of generic LDS ptr == LDS offset
    __builtin_amdgcn_global_load_async_to_lds_b128(g, l, 0, 0);
    __builtin_amdgcn_global_load_async_to_lds_b128(g, l, 16, 0);
}

// Double-buffered pipeline: prefetch row k+1 into LDS buffer (buf^1) via the
// async DMA path while dequantizing row k out of buffer buf. Each thread only
// touches the LDS bytes it itself requested, so the only sync needed is
// s_wait_asynccnt (no workgroup barriers).
__global__ __launch_bounds__(TPB) void dequant_gather_async(
    const float* __restrict__ weights, const float* __restrict__ scales,
    const int* __restrict__ tokens, float* __restrict__ out, int n) {
    __shared__ float lds[2][MODEL_DIM];               // 16 KB of the 320 KB/WGP pool
    const int t = threadIdx.x;
    const int stride = gridDim.x;
    int row = blockIdx.x;
    if (row >= n) return;

    int buf = 0;
    {   // prologue: start DMA of first row
        int tok = tokens[row];                        // uniform -> scalar load
        async_row_slice_to_lds(weights + (size_t)tok * MODEL_DIM + t * F_PER_T,
                               &lds[0][t * F_PER_T]);
    }
    for (; row < n; row += stride) {
        const int nrow = row + stride;
        const bool hasnext = nrow < n;
        if (hasnext) {
            int ntok = tokens[nrow];
            async_row_slice_to_lds(weights + (size_t)ntok * MODEL_DIM + t * F_PER_T,
                                   &lds[buf ^ 1][t * F_PER_T]);
        }
        // Current row's 2 async ops are the oldest outstanding; next row's 2 may remain.
        if (hasnext) wait_async_2(); else wait_async_0();
        asm volatile("" ::: "memory");                // compiler barrier: LDS now holds row data

        const float s = scales[tokens[row]];          // uniform -> scalar loads
        const v4f* l = (const v4f*)&lds[buf][t * F_PER_T];
        v4f a = l[0];
        v4f b = l[1];
        v4f* orow = (v4f*)(out + (size_t)row * MODEL_DIM) + 2 * t;
        __builtin_nontemporal_store(dq4(a, s), orow);     // streaming store, skip L2 retention
        __builtin_nontemporal_store(dq4(b, s), orow + 1);
        buf ^= 1;
    }
}
#endif  // USE_ASYNC_PATH

// Fallback: one row per block, direct NT b128 loads -> dequant -> NT b128 stores.
__global__ __launch_bounds__(TPB) void dequant_gather_direct(
    const float* __restrict__ weights, const float* __restrict__ scales,
    const int* __restrict__ tokens, float* __restrict__ out, int n) {
    const int row = blockIdx.x;
    if (row >= n) return;
    const int tok = tokens[row];                      // uniform -> scalar load
    const float s = scales[tok];                      // uniform -> scalar load
    const v4f* wrow = (const v4f*)(weights + (size_t)tok * MODEL_DIM);
    v4f* orow = (v4f*)(out + (size_t)row * MODEL_DIM);
    const int t = threadIdx.x;
    v4f a = __builtin_nontemporal_load(wrow + t);         // rows are use-once streams
    v4f b = __builtin_nontemporal_load(wrow + t + TPB);
    __builtin_nontemporal_store(dq4(a, s), orow + t);
    __builtin_nontemporal_store(dq4(b, s), orow + t + TPB);
}

extern "C" void kernel_launch(void* const* d_in, const int* in_sizes, int n_in,
                              void* d_out, int out_size, void* d_ws, size_t ws_size,
                              hipStream_t stream) {
    const float* weights = (const float*)d_in[0];   // [VOCAB, MODEL_DIM] f32
    const float* scales  = (const float*)d_in[1];   // [VOCAB] f32
    const int*   tokens  = (const int*)d_in[2];     // [N_TOKENS] i32
    float* out = (float*)d_out;                     // [N_TOKENS, MODEL_DIM] f32
    const int n = in_sizes[2];
    (void)d_ws; (void)ws_size; (void)n_in; (void)out_size;

#ifdef USE_ASYNC_PATH
    int grid = n < 2048 ? n : 2048;                 // 4-row pipeline per block at n=8192
    hipLaunchKernelGGL(dequant_gather_async, dim3(grid), dim3(TPB), 0, stream,
                       weights, scales, tokens, out, n);
#else
    hipLaunchKernelGGL(dequant_gather_direct, dim3(n), dim3(TPB), 0, stream,
                       weights, scales, tokens, out, n);
#endif
}